// MinkowskiBroadcast_54357106098853
// MI455X (gfx1250) — compile-verified
//
#include <hip/hip_runtime.h>
#include <stdint.h>

// MinkowskiBroadcast gather: out[i, :] = x_glob[batch_idx[i], :]
//   x_glob:    [NB=32, C=128] f32  (16 KB, staged in LDS via async copy)
//   batch_idx: [N=1e6] i32
//   out:       [N, 128] f32  (512 MB, pure store-bandwidth bound: ~22us @ 23.3 TB/s)
//
// CDNA5 paths used:
//  - global_load_async_to_lds_b128 + s_wait_asynccnt (ASYNCcnt) to fill LDS table
//  - ds_load_b128 row reads from LDS (DS pipe), freeing VMEM pipe for stores
//  - non-temporal global_store_b128 for the 512 MB write-once stream
//  - v_readlane broadcast of per-row index (1 coalesced index load per 32 rows)

typedef __attribute__((ext_vector_type(4))) float f32x4;

#define TILE_F4   (32 * 128 / 4)   // 1024 float4 = 16 KB
#define BLK       256              // 8 waves of 32

__global__ __launch_bounds__(BLK)
void mb_gather_kernel(const float* __restrict__ xg,
                      const int*   __restrict__ bidx,
                      f32x4*       __restrict__ out4,
                      int n, unsigned xg_bytes)
{
    __shared__ f32x4 tile[TILE_F4];

    const unsigned tid  = threadIdx.x;
    const unsigned lane = tid & 31u;
    const unsigned wib  = tid >> 5;       // wave-in-block, 0..7

    // ---- Stage x_glob into LDS with CDNA5 async global->LDS copies ----------
    // Generic shared pointer: low 32 bits are the LDS byte offset (aperture in
    // the high bits), which is what the async instruction's VDST expects.
    {
        unsigned    lds_base = (unsigned)(uintptr_t)(&tile[0]);
        const char* gsrc     = (const char*)xg;
        for (unsigned off = tid * 16u; off < xg_bytes; off += BLK * 16u) {
            unsigned    laddr = lds_base + off;
            const char* gaddr = gsrc + off;
            asm volatile("global_load_async_to_lds_b128 %0, %1, off"
                         :: "v"(laddr), "v"(gaddr) : "memory");
        }
        asm volatile("s_wait_asynccnt 0x0" ::: "memory");
        __syncthreads();
    }

    // ---- Stream rows: each wave copies 32 rows per chunk --------------------
    const long nchunks = ((long)n + 31) / 32;
    const long gwave   = (long)blockIdx.x * (BLK / 32) + wib;
    const long nwaves  = (long)gridDim.x * (BLK / 32);

    for (long chunk = gwave; chunk < nchunks; chunk += nwaves) {
        const long base = chunk * 32;
        const long rem  = (long)n - base;
        const int  cnt  = rem < 32 ? (int)rem : 32;

        // One coalesced b32 load of 32 indices per wave-chunk.
        int idx = 0;
        if ((int)lane < cnt)
            idx = __builtin_nontemporal_load(&bidx[base + lane]);

        // Lane L owns float4 column L of every row; row j adds j*512B, which
        // folds into the 24-bit store immediate under full unroll.
        f32x4* dst = out4 + base * 32 + lane;

        if (cnt == 32) {
#pragma unroll
            for (int j = 0; j < 32; ++j) {
                const int s = __builtin_amdgcn_readlane(idx, j);    // uniform row index
                const f32x4 v = tile[(unsigned)s * 32u + lane];     // ds_load_b128
                __builtin_nontemporal_store(v, dst + (long)j * 32); // global_store_b128 NT
            }
        } else {
            for (int j = 0; j < cnt; ++j) {
                const int s = __builtin_amdgcn_readlane(idx, j);
                const f32x4 v = tile[(unsigned)s * 32u + lane];
                __builtin_nontemporal_store(v, dst + (long)j * 32);
            }
        }
    }
}

extern "C" void kernel_launch(void* const* d_in, const int* in_sizes, int n_in,
                              void* d_out, int out_size, void* d_ws, size_t ws_size,
                              hipStream_t stream)
{
    const float* xg   = (const float*)d_in[0];   // [32, 128] f32
    const int*   bidx = (const int*)d_in[1];     // [N] i32
    f32x4*       out4 = (f32x4*)d_out;           // [N, 128] f32 viewed as [N*32] float4

    const int      n        = in_sizes[1];
    const unsigned xg_bytes = (unsigned)in_sizes[0] * 4u;

    // 1024 blocks * 8 waves = 8192 waves; ~3.8 chunks (of 32 rows) per wave.
    const int blocks = 1024;
    mb_gather_kernel<<<blocks, BLK, 0, stream>>>(xg, bidx, out4, n, xg_bytes);
}